// Char2Token2Mention_25555055411791
// MI455X (gfx1250) — compile-verified
//
#include <hip/hip_runtime.h>

typedef __attribute__((ext_vector_type(2))) float v2f;
typedef __attribute__((ext_vector_type(8))) float v8f;

#define D_DIM   256   // token feature dim (16 column tiles of 16)
#define TILE_M  16    // mentions per block
#define NWAVES  16    // one wave per 16-column tile -> 512 threads

// out[m, :] = sum_{j in segment(m)} spm_vals[j] * token_ft[token_code[j], :]
// spm_rows sorted  =>  segment(m0..m0+15) is one contiguous j-range.
// Tiled SpMM on v_wmma_f32_16x16x4_f32:  C(16x16) += A(16x4) x B(4x16)
//   A[m][k] = (spm_rows[j0+k] == m0+m) ? spm_vals[j0+k] : 0   (self-masking)
//   B[k][n] = token_ft[token_code[j0+k]][wave*16 + n]
__global__ __launch_bounds__(32 * NWAVES) void
mention_spmm_wmma(const float* __restrict__ token_ft,
                  const int*   __restrict__ token_code,
                  const int*   __restrict__ spm_rows,
                  const float* __restrict__ spm_vals,
                  float*       __restrict__ out,
                  int nnz)
{
    const int m0 = blockIdx.x * TILE_M;

    // --- per-block segment bounds via two binary searches (lower_bound) ---
    __shared__ int s_ptr[2];
    if (threadIdx.x < 2) {
        const int target = m0 + (threadIdx.x ? TILE_M : 0);
        int lo = 0, hi = nnz;
        while (lo < hi) {
            const int mid = (lo + hi) >> 1;
            if (spm_rows[mid] < target) lo = mid + 1; else hi = mid;
        }
        s_ptr[threadIdx.x] = lo;
    }
    __syncthreads();
    // Align start down to 4: j < rowptr[m0] has spm_rows[j] < m0 -> the
    // A-matrix compare rejects it, so extra leading elements contribute zero.
    const int jstart = s_ptr[0] & ~3;
    const int jend   = s_ptr[1];
    // Main-loop bound where a full aligned b128 chunk is in-bounds.
    // (nnz % 4 == 0 here, so jmain == jend and the tail below is dead.)
    const int nnz4  = nnz & ~3;
    const int jmain = (jend < nnz4) ? jend : nnz4;

    const int  lane   = threadIdx.x & 31;
    const int  wave   = threadIdx.x >> 5;        // column tile id, 0..15
    const int  nsub   = lane & 15;               // M (A lanes) / N (B,C lanes)
    const int  col    = wave * 16 + nsub;
    const bool hiHalf = lane >= 16;              // K offset +2 (A/B); M offset +8 (C)
    const int  myM    = m0 + nsub;

    v8f acc = {};   // 16x16 f32 C/D tile

    int j0 = jstart;
    if (j0 < jmain) {
        // Prime the software pipeline: chunk descriptors (uniform -> s_load_b128).
        int4   rr = *(const int4*  )(spm_rows   + j0);
        float4 vv = *(const float4*)(spm_vals   + j0);
        int4   cc = *(const int4*  )(token_code + j0);

        for (; j0 < jmain; j0 += 4) {
            // Issue B gathers first (depend only on already-resident cc).
            const int cA = hiHalf ? cc.z : cc.x;
            const int cB = hiHalf ? cc.w : cc.y;
            v2f b;
            b.x = token_ft[(size_t)cA * D_DIM + col];
            b.y = token_ft[(size_t)cB * D_DIM + col];

            // A tile from resident rr/vv.
            const int   rA = hiHalf ? rr.z : rr.x;
            const int   rB = hiHalf ? rr.w : rr.y;
            const float vA = hiHalf ? vv.z : vv.x;
            const float vB = hiHalf ? vv.w : vv.y;
            v2f a;
            a.x = (rA == myM) ? vA : 0.0f;
            a.y = (rB == myM) ? vB : 0.0f;

            // Prefetch next chunk descriptors (branch-free clamped index;
            // overlaps with the WMMA + B-load wait).
            const int jn = (j0 + 4 < jmain) ? (j0 + 4) : j0;
            const int4   rr_n = *(const int4*  )(spm_rows   + jn);
            const float4 vv_n = *(const float4*)(spm_vals   + jn);
            const int4   cc_n = *(const int4*  )(token_code + jn);

            // D = A x B + C  (8 args: neg_a, A, neg_b, B, c_mod, C, reuse_a, reuse_b)
            acc = __builtin_amdgcn_wmma_f32_16x16x4_f32(
                      false, a, false, b, (short)0, acc, false, false);

            rr = rr_n; vv = vv_n; cc = cc_n;
        }
    }

    // Tail: only reachable when nnz % 4 != 0 (at most one chunk).
    for (; j0 < jend; j0 += 4) {
        int r[4], c[4]; float v[4];
        #pragma unroll
        for (int k = 0; k < 4; ++k) {
            const bool ok = (j0 + k) < nnz;
            const int  jj = ok ? (j0 + k) : (nnz - 1);
            r[k] = spm_rows[jj];
            v[k] = ok ? spm_vals[jj] : 0.0f;   // zeroed val masks clamped row
            c[k] = token_code[jj];
        }
        const int   rA = hiHalf ? r[2] : r[0];
        const int   rB = hiHalf ? r[3] : r[1];
        const float vA = hiHalf ? v[2] : v[0];
        const float vB = hiHalf ? v[3] : v[1];
        const int   cA = hiHalf ? c[2] : c[0];
        const int   cB = hiHalf ? c[3] : c[1];
        v2f a, b;
        a.x = (rA == myM) ? vA : 0.0f;
        a.y = (rB == myM) ? vB : 0.0f;
        b.x = token_ft[(size_t)cA * D_DIM + col];
        b.y = token_ft[(size_t)cB * D_DIM + col];
        acc = __builtin_amdgcn_wmma_f32_16x16x4_f32(
                  false, a, false, b, (short)0, acc, false, false);
    }

    // C layout: VGPR p holds row M = p (lanes 0-15) / p+8 (lanes 16-31), N = lane&15.
    // Every output element written exactly once -> non-temporal stores.
    #pragma unroll
    for (int p = 0; p < 8; ++p) {
        const int row = m0 + p + (hiHalf ? 8 : 0);
        __builtin_nontemporal_store(acc[p], out + (size_t)row * D_DIM + col);
    }
}

extern "C" void kernel_launch(void* const* d_in, const int* in_sizes, int n_in,
                              void* d_out, int out_size, void* d_ws, size_t ws_size,
                              hipStream_t stream) {
    const float* token_ft   = (const float*)d_in[0];
    const int*   token_code = (const int*  )d_in[1];
    const int*   spm_rows   = (const int*  )d_in[2];
    const float* spm_vals   = (const float*)d_in[3];
    float*       out        = (float*)d_out;

    const int nnz        = in_sizes[1];
    const int n_mentions = out_size / D_DIM;
    const int grid       = n_mentions / TILE_M;   // 65536/16 = 4096 blocks

    hipLaunchKernelGGL(mention_spmm_wmma, dim3(grid), dim3(32 * NWAVES), 0, stream,
                       token_ft, token_code, spm_rows, spm_vals, out, nnz);
}